// iFFT_45183055954249
// MI455X (gfx1250) — compile-verified
//
#include <hip/hip_runtime.h>
#include <hip/hip_bf16.h>

// 128 complex 512x512 inverse FFTs with fftshift folded in as (-1)^k output
// signs. Each 512-pt IFFT = (W16 x M) .* T  x W32  via V_WMMA_F32_16X16X4_F32.
// Pass 1: rows, d_in -> d_out. Pass 2: columns, in place on d_out, applies
// 1/512^2 scaling. Real/imag are separate contiguous planes (2j, 2j+1).

typedef float v2f __attribute__((ext_vector_type(2)));
typedef float v8f __attribute__((ext_vector_type(8)));

__device__ __forceinline__ v8f wmma_f32(v2f a, v2f b, v8f c) {
  // (neg_a, A, neg_b, B, c_mod, C, reuse_a, reuse_b)
  return __builtin_amdgcn_wmma_f32_16x16x4_f32(false, a, false, b, (short)0, c,
                                               false, false);
}

#define PI2F 6.28318530717958647692f
// 16 vectors x 512 samples per plane; XOR swizzle keeps lane-strided-by-b
// accesses on distinct banks while preserving 16B contiguity for float4 I/O.
#define LDSIDX(b, s) (((b) << 9) + ((s) ^ ((b) << 2)))

// 16 complex length-512 vectors resident in sr/si (swizzled); replaced
// in place by their unnormalized inverse DFTs times scale * (-1)^k.
__device__ __forceinline__ void ifft512_tile(float* __restrict__ sr,
                                             float* __restrict__ si,
                                             int tid, float scale) {
  const int wave = tid >> 5;
  const int lane = tid & 31;
  const int lo = lane & 15;   // A: m, B: n(batch), C/D: n(batch)
  const int hi = lane >> 4;

  // --- per-thread A-operand constants (IFFT sign: e^{+i theta}) ---
  // A(16x4) element held by (lane, reg r): m = lane&15, k = 4*slice + 2*hi + r
  v2f a1r[4], a1i[4];   // W16: w16^{m*k}
#pragma unroll
  for (int s = 0; s < 4; ++s) {
#pragma unroll
    for (int r = 0; r < 2; ++r) {
      int k = 4 * s + 2 * hi + r;  // n1
      float ang = (float)((lo * k) & 15) * (PI2F / 16.0f);
      float sv, cv; __sincosf(ang, &sv, &cv);
      a1r[s][r] = cv; a1i[s][r] = sv;
    }
  }
  v2f a2r[8], a2i[8];   // W32 half 0: k2 = lo, 8 K-slices of 4 over n2
#pragma unroll
  for (int s = 0; s < 8; ++s) {
#pragma unroll
    for (int r = 0; r < 2; ++r) {
      int n2 = 4 * s + 2 * hi + r;
      float ang = (float)((lo * n2) & 31) * (PI2F / 32.0f);
      float sv, cv; __sincosf(ang, &sv, &cv);
      a2r[s][r] = cv; a2i[s][r] = sv;
    }
  }

  // ---- step 1: A[k1,n2] = (sum_n1 x[32 n1 + n2] w16^{k1 n1}) * w512^{k1 n2}
  // wave w owns n2 in {w, w+8, w+16, w+24}; strictly in-place per n2 column.
#pragma unroll
  for (int t = 0; t < 4; ++t) {
    int n2 = wave + 8 * t;
    v8f zr = {0, 0, 0, 0, 0, 0, 0, 0};
    v8f zi = {0, 0, 0, 0, 0, 0, 0, 0};
#pragma unroll
    for (int s = 0; s < 4; ++s) {
      int n1 = 4 * s + 2 * hi;  // reg r adds +0/+1
      v2f br, bi;
      br[0] = sr[LDSIDX(lo, 32 * n1 + n2)];
      br[1] = sr[LDSIDX(lo, 32 * n1 + 32 + n2)];
      bi[0] = si[LDSIDX(lo, 32 * n1 + n2)];
      bi[1] = si[LDSIDX(lo, 32 * n1 + 32 + n2)];
      v2f bin = { -bi[0], -bi[1] };  // f32 WMMA has no A/B NEG -> negate B
      zr = wmma_f32(a1r[s], br, zr);
      zr = wmma_f32(a1i[s], bin, zr);
      zi = wmma_f32(a1r[s], bi, zi);
      zi = wmma_f32(a1i[s], br, zi);
    }
#pragma unroll
    for (int v = 0; v < 8; ++v) {
      int m = v + 8 * hi;  // k1 (C/D row)
      float ang = (float)((m * n2) & 511) * (PI2F / 512.0f);
      float sv, cv; __sincosf(ang, &sv, &cv);
      float xr = zr[v], xi = zi[v];
      int sidx = 32 * m + n2;
      sr[LDSIDX(lo, sidx)] = xr * cv - xi * sv;
      si[LDSIDX(lo, sidx)] = xr * sv + xi * cv;
    }
  }
  __syncthreads();

  // ---- step 2: X[k1 + 16 k2] = sum_n2 A[k1,n2] w32^{k2 n2}
  // wave w owns k1 in {2w, 2w+1}; results held in regs until global barrier.
  v8f zracc[2][2], ziacc[2][2];
#pragma unroll
  for (int q = 0; q < 2; ++q) {
    int k1 = 2 * wave + q;
    v8f zr0 = {0, 0, 0, 0, 0, 0, 0, 0};
    v8f zi0 = zr0, zr1 = zr0, zi1 = zr0;
#pragma unroll
    for (int s = 0; s < 8; ++s) {
      int n2 = 4 * s + 2 * hi;
      v2f br, bi;
      br[0] = sr[LDSIDX(lo, 32 * k1 + n2)];
      br[1] = sr[LDSIDX(lo, 32 * k1 + n2 + 1)];
      bi[0] = si[LDSIDX(lo, 32 * k1 + n2)];
      bi[1] = si[LDSIDX(lo, 32 * k1 + n2 + 1)];
      v2f bin = { -bi[0], -bi[1] };
      zr0 = wmma_f32(a2r[s], br, zr0);
      zr0 = wmma_f32(a2i[s], bin, zr0);
      zi0 = wmma_f32(a2r[s], bi, zi0);
      zi0 = wmma_f32(a2i[s], br, zi0);
      // half 1: w32^{(16+k2) n2} = (-1)^{n2} w32^{k2 n2}; parity(n2) == r
      v2f a2rh = { a2r[s][0], -a2r[s][1] };
      v2f a2ih = { a2i[s][0], -a2i[s][1] };
      zr1 = wmma_f32(a2rh, br, zr1);
      zr1 = wmma_f32(a2ih, bin, zr1);
      zi1 = wmma_f32(a2rh, bi, zi1);
      zi1 = wmma_f32(a2ih, br, zi1);
    }
    zracc[q][0] = zr0; ziacc[q][0] = zi0;
    zracc[q][1] = zr1; ziacc[q][1] = zi1;
  }
  __syncthreads();  // all step-2 reads of the shared tile are done

#pragma unroll
  for (int q = 0; q < 2; ++q) {
    int k1 = 2 * wave + q;
    // fftshift factor (-1)^k, and parity(k1 + 16 k2) == parity(k1)
    float sgn = (k1 & 1) ? -scale : scale;
#pragma unroll
    for (int h = 0; h < 2; ++h) {
#pragma unroll
      for (int v = 0; v < 8; ++v) {
        int k2 = h * 16 + v + 8 * hi;
        int k = k1 + 16 * k2;
        sr[LDSIDX(lo, k)] = zracc[q][h][v] * sgn;
        si[LDSIDX(lo, k)] = ziacc[q][h][v] * sgn;
      }
    }
  }
  __syncthreads();
}

// ---------------- pass 1: row IFFTs (contiguous dim), d_in -> d_out --------
__global__ __launch_bounds__(256) void ifft_rows_kernel(
    const float* __restrict__ in, float* __restrict__ out) {
  __shared__ float sm[2 * 16 * 512];  // 64 KB: re plane + im plane
  float* sr = sm;
  float* si = sm + 16 * 512;
  const int j = blockIdx.x;   // complex image 0..127 (planes 2j, 2j+1)
  const int rb = blockIdx.y;  // row block 0..31 (16 rows each)
  const int tid = threadIdx.x;

  const float* gre = in + (size_t)(2 * j) * 262144 + (size_t)rb * 16 * 512;
  const float* gim = gre + 262144;
#pragma unroll
  for (int i = 0; i < 8; ++i) {
    int f4 = i * 256 + tid;        // 0..2047 float4 slots (16 rows x 128)
    int b = f4 >> 7, c4 = f4 & 127;
    float4 vr = ((const float4*)gre)[b * 128 + c4];
    float4 vi = ((const float4*)gim)[b * 128 + c4];
    int s = (4 * c4) ^ (b << 2);   // swizzle preserves 16B alignment
    *(float4*)&sr[(b << 9) + s] = vr;
    *(float4*)&si[(b << 9) + s] = vi;
  }
  __syncthreads();

  ifft512_tile(sr, si, tid, 1.0f);  // sign only; scale applied in pass 2

  float* ore = out + (size_t)(2 * j) * 262144 + (size_t)rb * 16 * 512;
  float* oim = ore + 262144;
#pragma unroll
  for (int i = 0; i < 8; ++i) {
    int f4 = i * 256 + tid;
    int b = f4 >> 7, c4 = f4 & 127;
    int s = (4 * c4) ^ (b << 2);
    ((float4*)ore)[b * 128 + c4] = *(const float4*)&sr[(b << 9) + s];
    ((float4*)oim)[b * 128 + c4] = *(const float4*)&si[(b << 9) + s];
  }
}

// ---------------- pass 2: column IFFTs, in place on d_out ------------------
__global__ __launch_bounds__(256) void ifft_cols_kernel(float* __restrict__ data) {
  __shared__ float sm[2 * 16 * 512];
  float* sr = sm;
  float* si = sm + 16 * 512;
  const int j = blockIdx.x;   // complex image
  const int cb = blockIdx.y;  // column block 0..31 (16 columns each)
  const int tid = threadIdx.x;
  const int b = tid & 15;     // local column = LDS vector index

  float* gre = data + (size_t)(2 * j) * 262144 + cb * 16;
  float* gim = gre + 262144;
#pragma unroll
  for (int i = 0; i < 32; ++i) {
    int rr = i * 16 + (tid >> 4);  // sample index = row
    sr[(b << 9) + (rr ^ (b << 2))] = gre[(size_t)rr * 512 + b];
    si[(b << 9) + (rr ^ (b << 2))] = gim[(size_t)rr * 512 + b];
  }
  __syncthreads();

  ifft512_tile(sr, si, tid, 1.0f / 262144.0f);  // ifft2 backward norm

#pragma unroll
  for (int i = 0; i < 32; ++i) {
    int rr = i * 16 + (tid >> 4);
    gre[(size_t)rr * 512 + b] = sr[(b << 9) + (rr ^ (b << 2))];
    gim[(size_t)rr * 512 + b] = si[(b << 9) + (rr ^ (b << 2))];
  }
}

extern "C" void kernel_launch(void* const* d_in, const int* in_sizes, int n_in,
                              void* d_out, int out_size, void* d_ws,
                              size_t ws_size, hipStream_t stream) {
  (void)in_sizes; (void)n_in; (void)d_ws; (void)ws_size; (void)out_size;
  const float* x = (const float*)d_in[0];
  float* out = (float*)d_out;
  dim3 grid(128, 32);  // 128 complex images x 32 blocks of 16 rows/cols
  ifft_rows_kernel<<<grid, 256, 0, stream>>>(x, out);
  ifft_cols_kernel<<<grid, 256, 0, stream>>>(out);
}